// VAE_student_rbf_26499948216482
// MI455X (gfx1250) — compile-verified
//
#include <hip/hip_runtime.h>

typedef __attribute__((ext_vector_type(2))) float v2f;
typedef __attribute__((ext_vector_type(8))) float v8f;

#define H 100
#define HSTR 108          // 108 mod 64 = 44, gcd(44,64)=4 -> conflict-free A reads
#define WAVES 4
#define LOG_PI 1.14472988584940017f

// Wave-private LDS ordering is guaranteed by hardware (DS in-order per wave);
// this only stops the compiler from reordering across the point. Emits nothing.
#define WFENCE() do { __builtin_amdgcn_wave_barrier(); asm volatile("" ::: "memory"); } while (0)

__device__ __forceinline__ float softplus_f(float v) {
    return (v > 0.0f) ? (v + log1pf(expf(-v))) : log1pf(expf(v));
}

__device__ __forceinline__ float st_lp(float xv, float df, float loc, float scale) {
    float y = (xv - loc) / scale;
    return lgammaf(0.5f * (df + 1.0f)) - lgammaf(0.5f * df)
         - 0.5f * (logf(df) + LOG_PI) - logf(scale)
         - 0.5f * (df + 1.0f) * log1pf(y * y / df);
}

// Two 16-row tiles through Linear(2,100)->ReLU->Linear(100,2) with f32 WMMA.
// The two accumulator chains are independent -> interleaved to hide WMMA RAW.
__device__ __forceinline__ void mlp_pair(
    v2f a0, v2f a1,
    const v2f (&B1)[7], const float (&bias1)[7],
    const float* __restrict__ w2l, const float* __restrict__ b2l,
    float* hid0, float* hid1, int lane, v8f& d0, v8f& d1)
{
    const int n = lane & 15;
    const bool hi = lane >= 16;

    // ---- Layer 1: [16,4(pad)] @ [4(pad),100] + b1, ReLU -> LDS (both tiles) ----
    #pragma unroll
    for (int c = 0; c < 7; ++c) {
        const int col = c * 16 + n;
        const float bv = bias1[c];
        v8f acc0 = {bv, bv, bv, bv, bv, bv, bv, bv};
        v8f acc1 = acc0;
        acc0 = __builtin_amdgcn_wmma_f32_16x16x4_f32(false, a0, false, B1[c],
                                                     (short)0, acc0, false, false);
        acc1 = __builtin_amdgcn_wmma_f32_16x16x4_f32(false, a1, false, B1[c],
                                                     (short)0, acc1, false, false);
        #pragma unroll
        for (int i = 0; i < 8; ++i) {
            const int m = i + (hi ? 8 : 0);
            float h0 = acc0[i] > 0.0f ? acc0[i] : 0.0f;
            float h1 = acc1[i] > 0.0f ? acc1[i] : 0.0f;
            hid0[m * HSTR + col] = h0;
            hid1[m * HSTR + col] = h1;
        }
    }
    WFENCE();

    // ---- Layer 2: [16,100] @ [100,2(pad16)] + b2, 2x25 interleaved WMMAs ----
    const int n2 = (n < 2) ? n : 0;            // clamped index, no branchy loads
    const float tb2 = b2l[n2];
    const float bv2 = (n < 2) ? tb2 : 0.0f;
    v8f acc0 = {bv2, bv2, bv2, bv2, bv2, bv2, bv2, bv2};
    v8f acc1 = acc0;
    const int koff = hi ? 2 : 0;
    #pragma unroll
    for (int kc = 0; kc < 25; ++kc) {
        const int k0 = 4 * kc + koff;
        v2f ah0 = *(const v2f*)&hid0[n * HSTR + k0];
        v2f ah1 = *(const v2f*)&hid1[n * HSTR + k0];
        const float t0 = w2l[k0 * 2 + n2];
        const float t1 = w2l[(k0 + 1) * 2 + n2];
        v2f b;
        b.x = (n < 2) ? t0 : 0.0f;
        b.y = (n < 2) ? t1 : 0.0f;
        acc0 = __builtin_amdgcn_wmma_f32_16x16x4_f32(false, ah0, false, b,
                                                     (short)0, acc0, false, false);
        acc1 = __builtin_amdgcn_wmma_f32_16x16x4_f32(false, ah1, false, b,
                                                     (short)0, acc1, false, false);
    }
    WFENCE();
    d0 = acc0;
    d1 = acc1;
}

// Pull the two 16x2 heads (C/D layout) out to per-row scalars; every lane
// owns one of the 32 rows afterwards.
__device__ __forceinline__ void extract_pair(v8f d0, v8f d1, float* res, int lane,
                                             float& rA, float& rB) {
    const int n = lane & 15;
    const bool hi = lane >= 16;
    if (n < 2) {
        #pragma unroll
        for (int i = 0; i < 8; ++i) {
            const int m = i + (hi ? 8 : 0);
            res[m * 2 + n] = d0[i];
            res[32 + m * 2 + n] = d1[i];
        }
    }
    WFENCE();
    rA = res[lane * 2 + 0];
    rB = res[lane * 2 + 1];
    WFENCE();
}

extern "C" __global__ __launch_bounds__(128)
void vae_fused(const float* __restrict__ x, const float* __restrict__ eps,
               const float* __restrict__ emu_w1, const float* __restrict__ emu_b1,
               const float* __restrict__ emu_w2, const float* __restrict__ emu_b2,
               const float* __restrict__ estd_w1, const float* __restrict__ estd_b1,
               const float* __restrict__ estd_w2, const float* __restrict__ estd_b2,
               const float* __restrict__ dmu_w1, const float* __restrict__ dmu_b1,
               const float* __restrict__ dmu_w2, const float* __restrict__ dmu_b2,
               const float* __restrict__ ddf_w1, const float* __restrict__ ddf_b1,
               const float* __restrict__ ddf_w2, const float* __restrict__ ddf_b2,
               const float* __restrict__ Cc, const float* __restrict__ Ww,
               float* __restrict__ out, int Nrows)
{
    __shared__ __align__(16) float lds_hid[WAVES][2][16 * HSTR];
    __shared__ __align__(16) float lds_res[WAVES][64];
    __shared__ __align__(16) float lds_w2[4][200];
    __shared__ __align__(16) float lds_b2[4][2];

    const int lane = threadIdx.x & 31;
    const int wave = threadIdx.x >> 5;
    const int n = lane & 15;
    const bool hi = lane >= 16;

    // ---- Stage layer-2 weights into LDS once per block ----
    for (int i = threadIdx.x; i < 4 * 200; i += 128) {
        const int m = i / 200, j = i - m * 200;
        const float* p = (m == 0) ? emu_w2 : (m == 1) ? estd_w2 : (m == 2) ? dmu_w2 : ddf_w2;
        lds_w2[m][j] = p[j];
    }
    if (threadIdx.x < 8) {
        const int m = threadIdx.x >> 1, j = threadIdx.x & 1;
        const float* p = (m == 0) ? emu_b2 : (m == 1) ? estd_b2 : (m == 2) ? dmu_b2 : ddf_b2;
        lds_b2[m][j] = p[j];
    }
    __syncthreads();

    // ---- Hoist layer-1 B fragments + biases into registers (loop-invariant) ----
    v2f B1[4][7];
    float bias1[4][7];
    {
        const float* w1s[4] = {emu_w1, estd_w1, dmu_w1, ddf_w1};
        const float* b1s[4] = {emu_b1, estd_b1, dmu_b1, ddf_b1};
        #pragma unroll
        for (int m = 0; m < 4; ++m) {
            #pragma unroll
            for (int c = 0; c < 7; ++c) {
                const int col = c * 16 + n;
                const bool cv = col < H;
                const int colc = cv ? col : (H - 1);      // clamp: unconditional load
                const float t0 = w1s[m][colc];
                const float t1 = w1s[m][H + colc];
                const float tb = b1s[m][colc];
                B1[m][c].x = (!hi && cv) ? t0 : 0.0f;     // K=0 row; K=2/3 pad -> 0
                B1[m][c].y = (!hi && cv) ? t1 : 0.0f;     // K=1 row
                bias1[m][c] = cv ? tb : 0.0f;
            }
        }
    }

    float* hid0 = lds_hid[wave][0];
    float* hid1 = lds_hid[wave][1];
    float* res  = lds_res[wave];

    const int ntile = Nrows >> 5;                       // 32 rows per tile-pair
    const int waveId = blockIdx.x * WAVES + wave;
    const int nwaves = gridDim.x * WAVES;
    const long long N2 = (long long)Nrows * 2;
    float* o_xmu = out + 1;
    float* o_xsc = out + 1 + N2;
    float* o_z   = out + 1 + 2 * N2;
    float* o_zmu = out + 1 + 3 * N2;
    float* o_zst = out + 1 + 4 * N2;

    float elbo_acc = 0.0f;

    for (int t = waveId; t < ntile; t += nwaves) {
        const int own = t * 32 + lane;                  // each lane owns one row
        const float xA = x[own * 2 + 0];
        const float xB = x[own * 2 + 1];
        // tile0 A-frag: rows 0..15 in lanes 0..15; tile1 rows live in lanes 16..31
        const float xAo = __shfl(xA, n | 16, 32);
        const float xBo = __shfl(xB, n | 16, 32);
        v2f ax0, ax1;
        ax0.x = hi ? 0.0f : xA;  ax0.y = hi ? 0.0f : xB;
        ax1.x = hi ? 0.0f : xAo; ax1.y = hi ? 0.0f : xBo;

        v8f d0, d1;
        float zmuA, zmuB, tA, tB;
        mlp_pair(ax0, ax1, B1[0], bias1[0], lds_w2[0], lds_b2[0], hid0, hid1, lane, d0, d1);
        extract_pair(d0, d1, res, lane, zmuA, zmuB);
        mlp_pair(ax0, ax1, B1[1], bias1[1], lds_w2[1], lds_b2[1], hid0, hid1, lane, d0, d1);
        extract_pair(d0, d1, res, lane, tA, tB);

        const float zstdA = softplus_f(tA), zstdB = softplus_f(tB);
        const float sigA = sqrtf(zstdA),    sigB = sqrtf(zstdB);
        const float eA = eps[own * 2 + 0],  eB = eps[own * 2 + 1];
        const float zA = zmuA + sigA * eA;
        const float zB = zmuB + sigB * eB;

        const float zAo = __shfl(zA, n | 16, 32);
        const float zBo = __shfl(zB, n | 16, 32);
        v2f az0, az1;
        az0.x = hi ? 0.0f : zA;  az0.y = hi ? 0.0f : zB;
        az1.x = hi ? 0.0f : zAo; az1.y = hi ? 0.0f : zBo;

        float xmuA, xmuB, uA, uB;
        mlp_pair(az0, az1, B1[2], bias1[2], lds_w2[2], lds_b2[2], hid0, hid1, lane, d0, d1);
        extract_pair(d0, d1, res, lane, xmuA, xmuB);
        mlp_pair(az0, az1, B1[3], bias1[3], lds_w2[3], lds_b2[3], hid0, hid1, lane, d0, d1);
        extract_pair(d0, d1, res, lane, uA, uB);
        const float dfA = softplus_f(uA), dfB = softplus_f(uB);

        // ---- RBF scale: exp(-5*d(z,C)) @ max(W,0) + 1e-10; scale = 1/inv ----
        const float zz = zA * zA + zB * zB;
        float invA = 1e-10f, invB = 1e-10f;
        for (int k = 0; k < 30; ++k) {
            const float c0 = Cc[k * 2 + 0], c1 = Cc[k * 2 + 1];
            const float dk = zz + c0 * c0 + c1 * c1 - 2.0f * (zA * c0 + zB * c1);
            const float ek = expf(-5.0f * dk);
            float w0 = Ww[k * 2 + 0];  w0 = w0 > 0.0f ? w0 : 0.0f;
            float w1v = Ww[k * 2 + 1]; w1v = w1v > 0.0f ? w1v : 0.0f;
            invA = fmaf(ek, w0, invA);
            invB = fmaf(ek, w1v, invB);
        }
        const float scA = 1.0f / invA, scB = 1.0f / invB;

        // ---- Student-t log p(x|z) and KL = logN(z;mu,sig) - logN(z;0,1) ----
        const float lp = st_lp(xA, dfA, xmuA, scA) + st_lp(xB, dfB, xmuB, scB);
        const float kl = (0.5f * (zA * zA - eA * eA) - logf(sigA))
                       + (0.5f * (zB * zB - eB * eB) - logf(sigB));
        elbo_acc += lp - kl;

        const long long idx = (long long)own * 2;
        o_xmu[idx] = xmuA;  o_xmu[idx + 1] = xmuB;
        o_xsc[idx] = scA;   o_xsc[idx + 1] = scB;
        o_z[idx]   = zA;    o_z[idx + 1]   = zB;
        o_zmu[idx] = zmuA;  o_zmu[idx + 1] = zmuB;
        o_zst[idx] = zstdA; o_zst[idx + 1] = zstdB;
    }

    // ---- One atomic per wave for the mean ----
    #pragma unroll
    for (int off = 16; off > 0; off >>= 1) elbo_acc += __shfl_xor(elbo_acc, off, 32);
    if (lane == 0) atomicAdd(out, elbo_acc * (1.0f / (float)Nrows));
}

extern "C" __global__ void vae_zero(float* out) {
    if (threadIdx.x == 0) out[0] = 0.0f;
}

extern "C" void kernel_launch(void* const* d_in, const int* in_sizes, int n_in,
                              void* d_out, int out_size, void* d_ws, size_t ws_size,
                              hipStream_t stream) {
    const float* x    = (const float*)d_in[0];
    const float* eps  = (const float*)d_in[1];
    const float* emu_w1  = (const float*)d_in[2];
    const float* emu_b1  = (const float*)d_in[3];
    const float* emu_w2  = (const float*)d_in[4];
    const float* emu_b2  = (const float*)d_in[5];
    const float* estd_w1 = (const float*)d_in[6];
    const float* estd_b1 = (const float*)d_in[7];
    const float* estd_w2 = (const float*)d_in[8];
    const float* estd_b2 = (const float*)d_in[9];
    const float* dmu_w1  = (const float*)d_in[10];
    const float* dmu_b1  = (const float*)d_in[11];
    const float* dmu_w2  = (const float*)d_in[12];
    const float* dmu_b2  = (const float*)d_in[13];
    const float* ddf_w1  = (const float*)d_in[14];
    const float* ddf_b1  = (const float*)d_in[15];
    const float* ddf_w2  = (const float*)d_in[16];
    const float* ddf_b2  = (const float*)d_in[17];
    const float* C       = (const float*)d_in[18];
    const float* W       = (const float*)d_in[19];
    float* out = (float*)d_out;

    const int Nrows = in_sizes[0] / 2;           // x is [N,2]
    const int ntile = Nrows / 32;                // 32 rows per wave-iteration
    int nblocks = 1024;                          // 4096 persistent waves
    if (nblocks * WAVES > ntile) nblocks = (ntile + WAVES - 1) / WAVES;

    vae_zero<<<1, 32, 0, stream>>>(out);
    vae_fused<<<nblocks, 128, 0, stream>>>(
        x, eps,
        emu_w1, emu_b1, emu_w2, emu_b2,
        estd_w1, estd_b1, estd_w2, estd_b2,
        dmu_w1, dmu_b1, dmu_w2, dmu_b2,
        ddf_w1, ddf_b1, ddf_w2, ddf_b2,
        C, W, out, Nrows);
}